// Recurrent_32478542692622
// MI455X (gfx1250) — compile-verified
//
#include <hip/hip_runtime.h>

#define STEPS 19
#define WAVES 4            // waves per block (128 threads)

typedef float v2f __attribute__((ext_vector_type(2)));
typedef float v8f __attribute__((ext_vector_type(8)));

// ---- per-wave LDS activation scratch (float offsets), bank-conflict-free strides ----
#define A_STRIDE  68        // actA: 16 x 68  (cols 0..49 data, 50 = bias-one, 51 = 0)
#define B_STRIDE  36        // actB: 16 x 36  (cols 0..19 data, 20 = bias-one, 21..23 = 0)
#define H_STRIDE  4         // actH: 16 x 4   (cols 0..1 h,    2  = bias-one, 3 = 0)
#define Z_STRIDE  (2*STEPS) // zt:   16 x 38
#define OFF_ACTA  0
#define OFF_ACTB  (16*A_STRIDE)                  // 1088
#define OFF_ACTH  (OFF_ACTB + 16*B_STRIDE)       // 1664
#define OFF_ZT    (OFF_ACTH + 16*H_STRIDE)       // 1728
#define PW_SIZE   (OFF_ZT + 16*Z_STRIDE)         // 2336 floats
#define SMEM_FLOATS (WAVES * PW_SIZE)            // 9344 floats = 37376 B

// fragment table offsets (v2f units): B-fragments kept in VGPRs for the whole kernel
#define FR_H1 0    // 4 tiles x 1 kstep
#define FR_H2 4    // 2 x 13
#define FR_H3 30   // 1 x 6
#define FR_Z1 36   // 4 x 1
#define FR_Z2 40   // 2 x 13
#define FR_Z3 66   // 1 x 6
#define FR_TOTAL 72

// padded (bias-augmented) weight element:
//   rows 0..K-1 = W, row K = bias, rows >K = 0; cols >= N = 0
__device__ __forceinline__ float welem(const float* W, const float* b,
                                       int K, int N, int r, int c) {
    if (c >= N) return 0.0f;
    if (r < K)  return W[r * N + c];
    if (r == K) return b[c];
    return 0.0f;
}

// Build B fragments (ISA 7.12.2 32-bit B layout, wave32):
//   vgpr j, lane l -> K = 4*k + 2*(l>>4) + j, N = 16*t + (l&15)
template <int K, int N, int KSTEPS, int NTILES>
__device__ __forceinline__ void load_frags(v2f* fr, const float* W,
                                           const float* bias, int lane) {
    const int half = lane >> 4, l15 = lane & 15;
#pragma unroll
    for (int t = 0; t < NTILES; ++t) {
#pragma unroll
        for (int k = 0; k < KSTEPS; ++k) {
            const int ka = 4 * k + 2 * half;
            const int c = t * 16 + l15;
            v2f b;
            b.x = welem(W, bias, K, N, ka + 0, c);
            b.y = welem(W, bias, K, N, ka + 1, c);
            fr[t * KSTEPS + k] = b;
        }
    }
}

__device__ __forceinline__ float sigmoid_f(float x) {
#if __has_builtin(__builtin_amdgcn_tanhf)
    // sigmoid(x) = 0.5 + 0.5*tanh(x/2) : mul + v_tanh_f32 (TRANS) + fma
    return __builtin_fmaf(0.5f, __builtin_amdgcn_tanhf(0.5f * x), 0.5f);
#else
    return __builtin_amdgcn_rcpf(1.0f + __expf(-x));
#endif
}

// One bias-augmented layer: out = act( [actIn | 1] @ [W ; bias] )
// A-fragments preloaded up front (batched ds_loads, staggered waits);
// k-outer / tile-inner so independent accumulator chains interleave in the XDL.
template <int KSTEPS, int NTILES, int MODE>
__device__ __forceinline__ void layerR(const float* actIn, int inStride,
                                       const v2f* bf, float* outP, int outStride,
                                       int outOff, int outLimit, int lane) {
    const int half = lane >> 4, l15 = lane & 15;
    v2f a[KSTEPS];
#pragma unroll
    for (int k = 0; k < KSTEPS; ++k)
        a[k] = *(const v2f*)(actIn + l15 * inStride + 4 * k + 2 * half);  // ds_load_b64

    v8f c[NTILES];
    const v8f zero = {0.f, 0.f, 0.f, 0.f, 0.f, 0.f, 0.f, 0.f};
#pragma unroll
    for (int t = 0; t < NTILES; ++t) c[t] = zero;
#pragma unroll
    for (int k = 0; k < KSTEPS; ++k) {
#pragma unroll
        for (int t = 0; t < NTILES; ++t)
            c[t] = __builtin_amdgcn_wmma_f32_16x16x4_f32(
                       false, a[k], false, bf[t * KSTEPS + k],
                       (short)0, c[t], false, false);
    }
#pragma unroll
    for (int t = 0; t < NTILES; ++t) {
        const int n = t * 16 + l15;
        if (n < outLimit) {
#pragma unroll
            for (int r = 0; r < 8; ++r) {
                float x = c[t][r];
                float v = (MODE == 0) ? fmaxf(x, 0.01f * x)   // leaky relu
                                      : sigmoid_f(x);         // sigmoid
                outP[(r + 8 * half) * outStride + outOff + n] = v;
            }
        }
    }
}

extern "C" __global__ __launch_bounds__(32 * WAVES)
void recnet_kernel(const float* __restrict__ w,
                   const float* __restrict__ gWh1, const float* __restrict__ gbh1,
                   const float* __restrict__ gWh2, const float* __restrict__ gbh2,
                   const float* __restrict__ gWh3, const float* __restrict__ gbh3,
                   const float* __restrict__ gWz1, const float* __restrict__ gbz1,
                   const float* __restrict__ gWz2, const float* __restrict__ gbz2,
                   const float* __restrict__ gWz3, const float* __restrict__ gbz3,
                   float* __restrict__ out) {
    __shared__ float smem[SMEM_FLOATS];
    const int tid = threadIdx.x;
    const int wave = tid >> 5, lane = tid & 31;

    // ---- all weight B-fragments into registers once (bias-augmented, zero-padded) ----
    v2f fr[FR_TOTAL];
    load_frags< 2, 50, 1, 4>(fr + FR_H1, gWh1, gbh1, lane);
    load_frags<50, 20, 13, 2>(fr + FR_H2, gWh2, gbh2, lane);
    load_frags<20,  2, 6, 1>(fr + FR_H3, gWh3, gbh3, lane);
    load_frags< 2, 50, 1, 4>(fr + FR_Z1, gWz1, gbz1, lane);
    load_frags<50, 20, 13, 2>(fr + FR_Z2, gWz2, gbz2, lane);
    load_frags<20,  2, 6, 1>(fr + FR_Z3, gWz3, gbz3, lane);

    float* actA = smem + wave * PW_SIZE + OFF_ACTA;
    float* actB = smem + wave * PW_SIZE + OFF_ACTB;
    float* actH = smem + wave * PW_SIZE + OFF_ACTH;
    float* zt   = smem + wave * PW_SIZE + OFF_ZT;

    const size_t b0 = ((size_t)blockIdx.x * WAVES + wave) * 16;

    // init h tile: cols 0..1 = w, col 2 = 1 (bias row selector), col 3 = 0
    for (int i = lane; i < 16 * H_STRIDE; i += 32) {
        int r = i >> 2, c2 = i & 3;
        actH[i] = (c2 < 2) ? w[(b0 + (size_t)r) * 2 + c2] : (c2 == 2 ? 1.0f : 0.0f);
    }
    // persistent pad columns (store limits below keep these intact across steps)
    for (int i = lane; i < 32; i += 32) {           // actA cols 50(=1), 51(=0)
        int r = i >> 1, c2 = i & 1;
        actA[r * A_STRIDE + 50 + c2] = (c2 == 0) ? 1.0f : 0.0f;
    }
    for (int i = lane; i < 64; i += 32) {           // actB cols 20(=1), 21..23(=0)
        int r = i >> 2, c2 = i & 3;
        actB[r * B_STRIDE + 20 + c2] = (c2 == 0) ? 1.0f : 0.0f;
    }

    for (int s = 0; s < STEPS; ++s) {
        // h-path (leaky relu)
        layerR< 1, 4, 0>(actH, H_STRIDE, fr + FR_H1, actA, A_STRIDE, 0, 50, lane);
        layerR<13, 2, 0>(actA, A_STRIDE, fr + FR_H2, actB, B_STRIDE, 0, 20, lane);
        layerR< 6, 1, 0>(actB, B_STRIDE, fr + FR_H3, actH, H_STRIDE, 0,  2, lane);
        // z-path (sigmoid) from updated h
        layerR< 1, 4, 1>(actH, H_STRIDE, fr + FR_Z1, actA, A_STRIDE, 0, 50, lane);
        layerR<13, 2, 1>(actA, A_STRIDE, fr + FR_Z2, actB, B_STRIDE, 0, 20, lane);
        layerR< 6, 1, 1>(actB, B_STRIDE, fr + FR_Z3, zt, Z_STRIDE, 2 * s, 2, lane);
    }

    // coalesced write of this wave's [16][19][2] tile (contiguous 608 floats)
    float* op = out + b0 * (size_t)(2 * STEPS);
    for (int i = lane; i < 16 * 2 * STEPS; i += 32) op[i] = zt[i];
}

extern "C" void kernel_launch(void* const* d_in, const int* in_sizes, int n_in,
                              void* d_out, int out_size, void* d_ws, size_t ws_size,
                              hipStream_t stream) {
    const float* w   = (const float*)d_in[0];
    const float* Wh1 = (const float*)d_in[1];
    const float* bh1 = (const float*)d_in[2];
    const float* Wh2 = (const float*)d_in[3];
    const float* bh2 = (const float*)d_in[4];
    const float* Wh3 = (const float*)d_in[5];
    const float* bh3 = (const float*)d_in[6];
    const float* Wz1 = (const float*)d_in[7];
    const float* bz1 = (const float*)d_in[8];
    const float* Wz2 = (const float*)d_in[9];
    const float* bz2 = (const float*)d_in[10];
    const float* Wz3 = (const float*)d_in[11];
    const float* bz3 = (const float*)d_in[12];
    float* out = (float*)d_out;

    const int B = in_sizes[0] / 2;               // 1048576 rows
    const int rowsPerBlock = 16 * WAVES;         // 64
    const int blocks = B / rowsPerBlock;         // 16384 (B divisible)

    recnet_kernel<<<blocks, 32 * WAVES, 0, stream>>>(
        w, Wh1, bh1, Wh2, bh2, Wh3, bh3, Wz1, bz1, Wz2, bz2, Wz3, bz3, out);
}